// Conf_MPULoss_67396626809312
// MI455X (gfx1250) — compile-verified
//
#include <hip/hip_runtime.h>

#define NQ    22      // number of scalar accumulators
#define EPSF  1e-9f
#define THRF  0.5f

typedef float v2f __attribute__((ext_vector_type(2)));
typedef float v8f __attribute__((ext_vector_type(8)));

__device__ __forceinline__ float wave_reduce_add(float v) {
#pragma unroll
    for (int o = 16; o; o >>= 1) v += __shfl_xor(v, o, 32);
    return v;
}

// Stage 1: stream x/t once, accumulate 22 masked sums per thread, reduce
// thread->wave via LDS atomics, waves->block via f32 WMMA (16x16x4, A=ones
// so D[m,n] = sum_k B[k,n] — exact f32 column sums), write per-block
// partials to ws in column-major [quantity][block] layout.
__global__ __launch_bounds__(256) void conf_mpu_stage1(
    const float* __restrict__ x, const int* __restrict__ t,
    float* __restrict__ ws, int N, int NB)
{
    __shared__ float sacc[8][32];     // [wave][quantity], padded to 32
    const int tid  = threadIdx.x;
    const int lane = tid & 31;
    const int wid  = tid >> 5;

    sacc[wid][lane] = 0.0f;
    __syncthreads();

    float acc[NQ];
#pragma unroll
    for (int j = 0; j < NQ; ++j) acc[j] = 0.0f;

    const long stride = (long)gridDim.x * blockDim.x;
    for (long i = (long)blockIdx.x * blockDim.x + tid; i < N; i += stride) {
        const float* row = x + 5 * i;
        // prefetch next tile for this lane (speculative: OOB is dropped)
        __builtin_prefetch(row + 5 * stride, 0, 3);

        const float x0 = row[0], x1 = row[1], x2 = row[2], x3 = row[3], x4 = row[4];
        const int   ti = t[i];

        const float m  = fmaxf(fmaxf(fmaxf(x0, x1), fmaxf(x2, x3)), x4);
        const float e0 = __expf(x0 - m), e1 = __expf(x1 - m), e2 = __expf(x2 - m);
        const float e3 = __expf(x3 - m), e4 = __expf(x4 - m);
        const float inv = 1.0f / (e0 + e1 + e2 + e3 + e4);
        const float p0 = e0 * inv, p1 = e1 * inv, p2 = e2 * inv;
        const float p3 = e3 * inv, p4 = e4 * inv;

        const float lpn = -__logf(p4 + EPSF);                    // -log(p_neg+eps)
        const float pt  = (ti == 0) ? p0 : (ti == 1) ? p1 :
                          (ti == 2) ? p2 : (ti == 3) ? p3 : p4;  // p[t]
        const float lpt = -__logf(pt + EPSF);                    // -log(p_t+eps)
        const float rr  = lpn / pt;                              // masked later

        const float pc[4] = {p0, p1, p2, p3};
#pragma unroll
        for (int c = 0; c < 5; ++c) acc[c] += (ti == c) ? 1.0f : 0.0f;   // counts
#pragma unroll
        for (int c = 0; c < 4; ++c) {
            const bool is = (ti == c);
            acc[5 + c]  += is ? lpt : 0.0f;                      // class_losses
            acc[9 + c]  += is ? lpn : 0.0f;                      // loss_neg
            const bool cm = is && (pc[c] > THRF);
            acc[13 + c] += cm ? 1.0f : 0.0f;                     // nsel
            acc[17 + c] += cm ? rr   : 0.0f;                     // conf li
        }
        const bool mn = (ti == 4) && (p0 <= THRF) && (p1 <= THRF) &&
                        (p2 <= THRF) && (p3 <= THRF) && (p4 <= THRF);
        acc[21] += mn ? lpn : 0.0f;                              // li_n
    }

    // thread -> per-wave partials (LDS float atomics, ds_add_f32)
#pragma unroll
    for (int j = 0; j < NQ; ++j) atomicAdd(&sacc[wid][j], acc[j]);
    __syncthreads();

    // waves -> block partials via exact f32 WMMA column sums.
    // B layout (4x16, 2 VGPRs): VGPR0 = row K=0 (lanes 0-15) / K=2 (16-31),
    //                           VGPR1 = row K=1 (lanes 0-15) / K=3 (16-31).
    // A = all ones (layout-independent). D[m,n] = sum_k B[k,n].
    if (wid == 0) {                    // uniform per-wave branch: EXEC all-ones
        const int  col = lane & 15;
        const bool hi  = lane >= 16;
        v2f A; A[0] = 1.0f; A[1] = 1.0f;
        v8f Z = {};

        // quantities 0..15
        v2f B;
        B[0] = hi ? sacc[2][col] : sacc[0][col];
        B[1] = hi ? sacc[3][col] : sacc[1][col];
        v8f d = __builtin_amdgcn_wmma_f32_16x16x4_f32(
            false, A, false, B, (short)0, Z, false, false);
        B[0] = hi ? sacc[6][col] : sacc[4][col];
        B[1] = hi ? sacc[7][col] : sacc[5][col];
        d = __builtin_amdgcn_wmma_f32_16x16x4_f32(
            false, A, false, B, (short)0, d, false, false);

        // quantities 16..21 (slots 22..31 stay zero)
        v2f B2;
        B2[0] = hi ? sacc[2][16 + col] : sacc[0][16 + col];
        B2[1] = hi ? sacc[3][16 + col] : sacc[1][16 + col];
        v8f d2 = __builtin_amdgcn_wmma_f32_16x16x4_f32(
            false, A, false, B2, (short)0, Z, false, false);
        B2[0] = hi ? sacc[6][16 + col] : sacc[4][16 + col];
        B2[1] = hi ? sacc[7][16 + col] : sacc[5][16 + col];
        d2 = __builtin_amdgcn_wmma_f32_16x16x4_f32(
            false, A, false, B2, (short)0, d2, false, false);

        // D VGPR0, lanes 0-15 = row M=0, cols 0-15 (all rows equal)
        if (!hi) {
            ws[(long)col * NB + blockIdx.x] = d[0];
            if (col < NQ - 16)
                ws[(long)(16 + col) * NB + blockIdx.x] = d2[0];
        }
    }
}

// Stage 2: one block reduces the NB per-block partials per quantity and
// evaluates the closed-form loss.
__global__ __launch_bounds__(256) void conf_mpu_finalize(
    const float* __restrict__ ws, int NB, int N, float* __restrict__ out)
{
    __shared__ float wsum[8];
    __shared__ float tot[NQ];
    const int tid  = threadIdx.x;
    const int lane = tid & 31;
    const int wid  = tid >> 5;

    for (int j = 0; j < NQ; ++j) {
        float v = 0.0f;
        for (int b = tid; b < NB; b += 256) v += ws[(long)j * NB + b];
        v = wave_reduce_add(v);
        if (lane == 0) wsum[wid] = v;
        __syncthreads();
        if (tid == 0) {
            float s = 0.0f;
#pragma unroll
            for (int w = 0; w < 8; ++w) s += wsum[w];
            tot[j] = s;
        }
        __syncthreads();
    }

    if (tid == 0) {
        const float Nf = (float)N;
        float r1 = 0.0f, r2 = 0.0f, r3 = 0.0f;
#pragma unroll
        for (int c = 0; c < 4; ++c) {
            const float cnt   = tot[c];
            const float prior = cnt / Nf;
            const float den   = fmaxf(1.0f, cnt);
            r1 += prior * tot[5 + c] / den;
            r2 += prior * (tot[17 + c] / fmaxf(tot[13 + c], 1.0f));
            r3 += prior * tot[9 + c] / den;
        }
        const float r4 = tot[21] / fmaxf(tot[4], 1.0f);
        float pr = 4.0f * (r1 + r2 - r3);
        if (pr < 0.0f) pr = 0.0f;
        out[0] = pr + r4;
    }
}

extern "C" void kernel_launch(void* const* d_in, const int* in_sizes, int n_in,
                              void* d_out, int out_size, void* d_ws, size_t ws_size,
                              hipStream_t stream) {
    const float* x = (const float*)d_in[0];
    const int*   t = (const int*)d_in[1];
    float*     out = (float*)d_out;
    float*      ws = (float*)d_ws;
    const int N = in_sizes[1];            // t has N elements

    int NB = 2048;
    if ((size_t)NQ * NB * sizeof(float) > ws_size) {
        NB = (int)(ws_size / (NQ * sizeof(float)));
        if (NB < 1) NB = 1;
    }

    conf_mpu_stage1<<<NB, 256, 0, stream>>>(x, t, ws, N, NB);
    conf_mpu_finalize<<<1, 256, 0, stream>>>(ws, NB, N, out);
}